// PagedTransformerBlock_59914793779325
// MI455X (gfx1250) — compile-verified
//
#include <hip/hip_runtime.h>
#include <hip/hip_bf16.h>
#include <stdint.h>

// ---------------------------------------------------------------------------
// PagedTransformerBlock decode step for gfx1250 (MI455X).
// B=64, D=4096, H=32, HD=128, S=512, BS=16, MBS=32, DFF=16384.
// ~2.1 GB of weight+KV traffic per call -> bandwidth bound at 23.3 TB/s.
// Skinny (M=64) fp32 GEMMs use V_WMMA_F32_16X16X4_F32; LDS staging uses the
// CDNA5 async path (GLOBAL_LOAD_ASYNC_TO_LDS_B128 + s_wait_asynccnt).
// ---------------------------------------------------------------------------

typedef float v2f __attribute__((ext_vector_type(2)));
typedef float v8f __attribute__((ext_vector_type(8)));

#define BQ   64        // batch (GEMM M)
#define DIM  4096
#define NH   32
#define HD   128
#define SMAX 512
#define PBS  16        // paged block size
#define MBS  32        // blocks per sequence
#define NEGINF (-1e30f)
// LDS row stride in floats: 36*4 = 144 bytes -> 16B aligned rows (async B128),
// and r*36 mod 64 is distinct for r = 0..15 -> conflict-free fragment reads.
#define LSTR 36

// -------------------- CDNA5 async memory->LDS helpers ----------------------
__device__ __forceinline__ void async_load_b128(uint32_t lds_off, const float* gptr)
{
    asm volatile("global_load_async_to_lds_b128 %0, %1, off"
                 :: "v"(lds_off), "v"((uint64_t)(uintptr_t)gptr)
                 : "memory");
}
__device__ __forceinline__ void wait_async0()
{
    asm volatile("s_wait_asynccnt 0x0" ::: "memory");
}
__device__ __forceinline__ uint32_t lds_off_of(const float* p)
{
    return (uint32_t)(uintptr_t)p;   // low 32 bits of shared-aperture address
}

// ---------------------------------------------------------------------------
// RMSNorm: one block per batch row.
// ---------------------------------------------------------------------------
__global__ __launch_bounds__(256) void rmsnorm_kernel(
    const float* __restrict__ x, const float* __restrict__ g,
    float* __restrict__ out)
{
    const int b   = blockIdx.x;
    const int tid = threadIdx.x;
    const float* xr = x + (size_t)b * DIM;

    float ss = 0.f;
    for (int i = tid; i < DIM; i += 256) { float v = xr[i]; ss += v * v; }

    __shared__ float red[256];
    red[tid] = ss;
    __syncthreads();
    for (int s = 128; s > 0; s >>= 1) {
        if (tid < s) red[tid] += red[tid + s];
        __syncthreads();
    }
    const float inv = rsqrtf(red[0] / (float)DIM + 1e-5f);

    float* orow = out + (size_t)b * DIM;
    for (int i = tid; i < DIM; i += 256) orow[i] = xr[i] * inv * g[i];
}

// ---------------------------------------------------------------------------
// WMMA tile body: C[64,N](tile at blockIdx.x*64) = A[64,kbeg:kend] @ W^T chunk.
// Raw store (epilogues are handled by reduce/elementwise kernels).
// Block: 256 threads = 8 waves; 64x64 tile; wave w: N subtile w&3,
// M subtiles (w>>2)*2 + {0,1}. K chunk = 32, staged into LDS via async loads.
// Fragment layouts per CDNA5 ISA 7.12.2 (f32 16x16x4).
// ---------------------------------------------------------------------------
__device__ __forceinline__ void gemm_tile_body(
    const float* __restrict__ A, const float* __restrict__ W,
    float* __restrict__ C, float* lh, float* lw,
    int N, int K, int kbeg, int kend)
{
    const int tid   = threadIdx.x;
    const int lane  = tid & 31;
    const int wave  = tid >> 5;
    const int l15   = lane & 15;
    const int khalf = lane >> 4;          // 0: K0/K1, 1: K2/K3
    const int nt    = wave & 3;
    const int mt0   = (wave >> 2) * 2;
    const int nblk  = blockIdx.x * 64;

    // per-thread staging coordinates (2 float4 per array per chunk)
    const int srow0 = tid >> 3;                 // 0..31
    const int sc4   = (tid & 7) * 4;            // 0,4,..,28

    v8f acc0 = {}; v8f acc1 = {};

    for (int k0 = kbeg; k0 < kend; k0 += 32) {
        // --- async stage A chunk (64 rows x 32 K) ---
        async_load_b128(lds_off_of(&lh[srow0 * LSTR + sc4]),
                        A + (size_t)srow0 * K + k0 + sc4);
        async_load_b128(lds_off_of(&lh[(srow0 + 32) * LSTR + sc4]),
                        A + (size_t)(srow0 + 32) * K + k0 + sc4);
        // --- async stage W chunk (64 rows x 32 K) ---
        async_load_b128(lds_off_of(&lw[srow0 * LSTR + sc4]),
                        W + (size_t)(nblk + srow0) * K + k0 + sc4);
        async_load_b128(lds_off_of(&lw[(srow0 + 32) * LSTR + sc4]),
                        W + (size_t)(nblk + srow0 + 32) * K + k0 + sc4);
        // prefetch next W chunk
        if (k0 + 32 < kend)
            __builtin_prefetch(W + (size_t)(nblk + srow0) * K + k0 + 32 + sc4, 0, 1);

        wait_async0();
        __syncthreads();

        #pragma unroll
        for (int kk = 0; kk < 8; ++kk) {
            const int klo = kk * 4 + khalf * 2;
            const v2f bfrag = *(const v2f*)(&lw[(nt * 16 + l15) * LSTR + klo]);
            const v2f a0 = *(const v2f*)(&lh[(mt0 * 16 + l15) * LSTR + klo]);
            acc0 = __builtin_amdgcn_wmma_f32_16x16x4_f32(
                false, a0, false, bfrag, (short)0, acc0, false, false);
            const v2f a1 = *(const v2f*)(&lh[((mt0 + 1) * 16 + l15) * LSTR + klo]);
            acc1 = __builtin_amdgcn_wmma_f32_16x16x4_f32(
                false, a1, false, bfrag, (short)0, acc1, false, false);
        }
        __syncthreads();
    }

    const int col = nblk + nt * 16 + l15;
    #pragma unroll
    for (int t = 0; t < 2; ++t) {
        const v8f acc = t ? acc1 : acc0;
        const int mb = (mt0 + t) * 16;
        #pragma unroll
        for (int r = 0; r < 8; ++r) {
            const int row = mb + r + khalf * 8;
            C[(size_t)row * N + col] = acc[r];
        }
    }
}

// Fused multi-weight GEMM: blockIdx.y selects (W, C) pair. Raw store.
__global__ __launch_bounds__(256) void gemm_multi_kernel(
    const float* __restrict__ A,
    const float* __restrict__ Wa, const float* __restrict__ Wb,
    const float* __restrict__ Wc,
    float* __restrict__ Ca, float* __restrict__ Cb, float* __restrict__ Cc,
    int N, int K)
{
    __shared__ __align__(16) float lh[BQ * LSTR];
    __shared__ __align__(16) float lw[BQ * LSTR];
    const float* Ws[3] = { Wa, Wb, Wc };
    float*       Cs[3] = { Ca, Cb, Cc };
    gemm_tile_body(A, Ws[blockIdx.y], Cs[blockIdx.y], lh, lw, N, K, 0, K);
}

// Split-K GEMM: blockIdx.y = K-split index; raw partials into part buffer.
__global__ __launch_bounds__(256) void gemm_splitk_kernel(
    const float* __restrict__ A, const float* __restrict__ W,
    float* __restrict__ part, int N, int K, int kchunk)
{
    __shared__ __align__(16) float lh[BQ * LSTR];
    __shared__ __align__(16) float lw[BQ * LSTR];
    const int ks = blockIdx.y;
    gemm_tile_body(A, W, part + (size_t)ks * BQ * N, lh, lw,
                   N, K, ks * kchunk, (ks + 1) * kchunk);
}

// out = sum of 4 split-K partials (+ residual)
__global__ __launch_bounds__(256) void reduce4_kernel(
    const float* __restrict__ part, const float* __restrict__ resid,
    float* __restrict__ out, int N)
{
    const size_t i  = (size_t)blockIdx.x * 256 + threadIdx.x;
    const size_t MN = (size_t)BQ * N;
    float v = part[i] + part[i + MN] + part[i + 2 * MN] + part[i + 3 * MN];
    if (resid) v += resid[i];
    out[i] = v;
}

// t1 <- silu(t1) * t3   (elementwise SwiGLU gate)
__global__ __launch_bounds__(256) void swiglu_kernel(
    float* __restrict__ t1, const float* __restrict__ t3)
{
    const size_t i = (size_t)blockIdx.x * 256 + threadIdx.x;
    const float g = t1[i];
    t1[i] = (g / (1.f + __expf(-g))) * t3[i];
}

// ---------------------------------------------------------------------------
// Paged attention (decode): one block per (head, batch). 128 threads = 4 waves.
// The freshly projected k/v for this token replace the heap entry at s=ctx-1
// (functional equivalent of reshape_and_cache given block_table = arange).
// ---------------------------------------------------------------------------
__global__ __launch_bounds__(128) void paged_attn_kernel(
    const float* __restrict__ q, const float* __restrict__ knew,
    const float* __restrict__ vnew,
    const float* __restrict__ key_heap, const float* __restrict__ val_heap,
    const int* __restrict__ block_table, const int* __restrict__ ctx_lens,
    const float* __restrict__ scale_p, float* __restrict__ out)
{
    const int h    = blockIdx.x;
    const int b    = blockIdx.y;
    const int tid  = threadIdx.x;
    const int lane = tid & 31;
    const int wave = tid >> 5;
    const int ctx  = ctx_lens[b];
    const float scale = scale_p[0];

    __shared__ float sc[SMAX];
    __shared__ float qs[HD];
    __shared__ float red[128];

    qs[tid] = q[(size_t)b * DIM + h * HD + tid];
    __syncthreads();
    const float q0 = qs[lane], q1 = qs[lane + 32],
                q2 = qs[lane + 64], q3 = qs[lane + 96];

    for (int s = wave; s < SMAX; s += 4) {
        float sv = NEGINF;
        if (s < ctx) {
            const float* kp;
            if (s == ctx - 1) {
                kp = knew + (size_t)b * DIM + h * HD;
            } else {
                const int blk = block_table[b * MBS + (s >> 4)];
                kp = key_heap + (((size_t)blk * NH + h) * PBS + (s & 15)) * HD;
            }
            float p = q0 * kp[lane] + q1 * kp[lane + 32] +
                      q2 * kp[lane + 64] + q3 * kp[lane + 96];
            #pragma unroll
            for (int m = 16; m >= 1; m >>= 1) p += __shfl_xor(p, m, 32);
            sv = p * scale;
        }
        if (lane == 0) sc[s] = sv;
    }
    __syncthreads();

    float mx = fmaxf(fmaxf(sc[tid], sc[tid + 128]),
                     fmaxf(sc[tid + 256], sc[tid + 384]));
    red[tid] = mx;
    __syncthreads();
    for (int s2 = 64; s2 > 0; s2 >>= 1) {
        if (tid < s2) red[tid] = fmaxf(red[tid], red[tid + s2]);
        __syncthreads();
    }
    mx = red[0];
    __syncthreads();

    float psum = 0.f;
    #pragma unroll
    for (int i = 0; i < 4; ++i) {
        const int s = tid + i * 128;
        const float e = __expf(sc[s] - mx);
        sc[s] = e;
        psum += e;
    }
    red[tid] = psum;
    __syncthreads();
    for (int s2 = 64; s2 > 0; s2 >>= 1) {
        if (tid < s2) red[tid] += red[tid + s2];
        __syncthreads();
    }
    const float inv = 1.0f / red[0];

    float acc = 0.f;
    for (int s = 0; s < ctx; ++s) {
        const float* vp;
        if (s == ctx - 1) {
            vp = vnew + (size_t)b * DIM + h * HD;
        } else {
            const int blk = block_table[b * MBS + (s >> 4)];
            vp = val_heap + (((size_t)blk * NH + h) * PBS + (s & 15)) * HD;
        }
        acc += sc[s] * vp[tid];
    }
    out[(size_t)b * DIM + h * HD + tid] = acc * inv;
}

// ---------------------------------------------------------------------------
// Launch chain
// ---------------------------------------------------------------------------
extern "C" void kernel_launch(void* const* d_in, const int* in_sizes, int n_in,
                              void* d_out, int out_size, void* d_ws, size_t ws_size,
                              hipStream_t stream)
{
    const float* x         = (const float*)d_in[0];
    const float* key_heap  = (const float*)d_in[1];
    const float* val_heap  = (const float*)d_in[2];
    const int*   btab      = (const int*)d_in[3];
    // d_in[4] slot_mapping (int64) unused: derivable from context_lens
    const int*   ctx_lens  = (const int*)d_in[5];
    const float* scale     = (const float*)d_in[9];
    const float* Wq        = (const float*)d_in[13];
    const float* Wk        = (const float*)d_in[14];
    const float* Wv        = (const float*)d_in[15];
    const float* Wo        = (const float*)d_in[16];
    const float* W1        = (const float*)d_in[17];
    const float* W2        = (const float*)d_in[18];
    const float* W3        = (const float*)d_in[19];
    const float* g1        = (const float*)d_in[20];
    const float* g2        = (const float*)d_in[21];

    float* ws  = (float*)d_ws;
    const size_t RD = (size_t)BQ * DIM;      // 262144
    const size_t RF = (size_t)BQ * 4 * DIM;  // 1048576
    float* h    = ws;
    float* q    = h    + RD;
    float* k    = q    + RD;
    float* v    = k    + RD;
    float* ao   = v    + RD;   // attention output
    float* x2   = ao   + RD;   // post-Wo residual stream
    float* h2   = x2   + RD;
    float* t1   = h2   + RD;   // h2 @ W1^T  (later overwritten by swiglu)
    float* t3   = t1   + RF;   // h2 @ W3^T
    float* part = t3   + RF;   // split-K partials: 4 * 64 * 4096
    float* y    = (float*)d_out;

    // 1) h = rmsnorm(x, g1)
    rmsnorm_kernel<<<dim3(BQ), dim3(256), 0, stream>>>(x, g1, h);

    // 2) q,k,v = h @ {Wq,Wk,Wv}^T  -- fused, 192 blocks
    gemm_multi_kernel<<<dim3(DIM / 64, 3), dim3(256), 0, stream>>>(
        h, Wq, Wk, Wv, q, k, v, DIM, DIM);

    // 3) paged attention (new token's k/v substituted at s = ctx-1)
    paged_attn_kernel<<<dim3(NH, BQ), dim3(128), 0, stream>>>(
        q, k, v, key_heap, val_heap, btab, ctx_lens, scale, ao);

    // 4) x2 = ao @ Wo^T + x   -- split-K x4 (256 blocks) + reduce
    gemm_splitk_kernel<<<dim3(DIM / 64, 4), dim3(256), 0, stream>>>(
        ao, Wo, part, DIM, DIM, DIM / 4);
    reduce4_kernel<<<dim3((BQ * DIM) / 256), dim3(256), 0, stream>>>(
        part, x, x2, DIM);

    // 5) h2 = rmsnorm(x2, g2)
    rmsnorm_kernel<<<dim3(BQ), dim3(256), 0, stream>>>(x2, g2, h2);

    // 6) t1 = h2@W1^T, t3 = h2@W3^T  -- fused, 512 blocks
    gemm_multi_kernel<<<dim3((4 * DIM) / 64, 2), dim3(256), 0, stream>>>(
        h2, W1, W3, W3, t1, t3, t3, 4 * DIM, DIM);
    // t1 <- silu(t1) * t3
    swiglu_kernel<<<dim3((BQ * 4 * DIM) / 256), dim3(256), 0, stream>>>(t1, t3);

    // 7) y = t1 @ W2^T + x2  -- split-K x4 over K=16384 + reduce
    gemm_splitk_kernel<<<dim3(DIM / 64, 4), dim3(256), 0, stream>>>(
        t1, W2, part, DIM, 4 * DIM, DIM);
    reduce4_kernel<<<dim3((BQ * DIM) / 256), dim3(256), 0, stream>>>(
        part, x2, y, DIM);
}